// ActQuantWrapper_61598420959615
// MI455X (gfx1250) — compile-verified
//
#include <hip/hip_runtime.h>

// Problem dims (fixed by reference): x [4,2048,4096] -> M=8192 tokens, K=4096, N=4096.
#define M_TOK 8192
#define K_DIM 4096
#define N_DIM 4096

typedef __attribute__((ext_vector_type(2))) float v2f;
typedef __attribute__((ext_vector_type(8))) float v8f;

static constexpr int BM = 64;     // block tile M
static constexpr int BN = 128;    // block tile N
static constexpr int BK = 32;     // K step
static constexpr int PITCH = 36;  // floats; 36*4=144B rows -> every b128 dest 16B aligned

// ---------------------------------------------------------------------------
// CDNA5 async global->LDS copy (ASYNCcnt-tracked, no VGPR round trip).
// VDST operand = LDS byte offset (low 32 bits of the generic shared address;
// per the ISA's aperture rules).
// ---------------------------------------------------------------------------
__device__ __forceinline__ void async_copy_b128(void* lds_dst, const void* gsrc) {
    const uint32_t lds_off = (uint32_t)(uintptr_t)lds_dst;
    asm volatile("global_load_async_to_lds_b128 %0, %1, off"
                 :: "v"(lds_off), "v"(gsrc) : "memory");
}
__device__ __forceinline__ void wait_async0() {
    asm volatile("s_wait_asynccnt 0x0" ::: "memory");
}

// ---------------------------------------------------------------------------
// Kernel 1: per-token quant scale  s = max(absmax(x_row), 1e-5) / 127
// ---------------------------------------------------------------------------
__global__ void row_scale_kernel(const float* __restrict__ x,
                                 float* __restrict__ scales) {
    const int row = blockIdx.x;
    const float* xr = x + (size_t)row * K_DIM;
    __shared__ float red[256];
    float m = 0.0f;
    for (int i = threadIdx.x; i < K_DIM; i += 256) m = fmaxf(m, fabsf(xr[i]));
    red[threadIdx.x] = m;
    __syncthreads();
    #pragma unroll
    for (int s = 128; s > 0; s >>= 1) {
        if (threadIdx.x < s) red[threadIdx.x] = fmaxf(red[threadIdx.x], red[threadIdx.x + s]);
        __syncthreads();
    }
    if (threadIdx.x == 0) scales[row] = fmaxf(red[0], 1e-5f) * (1.0f / 127.0f);
}

// ---------------------------------------------------------------------------
// Kernel 2: y = fakequant(x) @ W^T + b   via V_WMMA_F32_16X16X4_F32
//   - 256 threads (8 wave32s as 2(M) x 4(N)), block tile 64x128, BK=32
//   - double-buffered LDS staged entirely by global_load_async_to_lds_b128
//   - per-token fake quant folded into A-fragment reads (scales preloaded)
// ---------------------------------------------------------------------------
__global__ void __launch_bounds__(256)
gemm_wmma_f32(const float* __restrict__ x, const float* __restrict__ scales,
              const float* __restrict__ W, const float* __restrict__ bias,
              float* __restrict__ y) {
    __shared__ __align__(16) float As[2][BM][PITCH];  // raw x tile, [m][k]
    __shared__ __align__(16) float Bs[2][BN][PITCH];  // W tile, [n][k] (K contiguous)

    const int tid  = threadIdx.x;
    const int m0   = blockIdx.y * BM;
    const int n0   = blockIdx.x * BN;
    const int wave = tid >> 5;
    const int lane = tid & 31;
    const int lhalf = lane >> 4;      // 0..1
    const int lmod  = lane & 15;      // 0..15
    const int wm = (wave & 1) * 32;   // wave M offset inside block tile
    const int wn = (wave >> 1) * 32;  // wave N offset inside block tile

    // Cooperative async-copy coordinates: thread t -> row (t>>3), float4 col (t&7)*4.
    const int lr = tid >> 3;          // 0..31
    const int lc = (tid & 7) * 4;     // 0,4,...,28

    // Loop-invariant per-token scales for the A rows this lane owns as M index.
    const float s0 = scales[m0 + wm + lmod];
    const float s1 = scales[m0 + wm + 16 + lmod];
    const float r0 = 1.0f / s0;
    const float r1 = 1.0f / s1;

    v8f acc[2][2] = {};

    auto issue_stage = [&](int kt, int buf) {
        #pragma unroll
        for (int p = 0; p < 2; ++p) {                 // A: 64x32 floats = 512 b128
            const int row = lr + 32 * p;
            async_copy_b128(&As[buf][row][lc],
                            x + (size_t)(m0 + row) * K_DIM + kt + lc);
        }
        #pragma unroll
        for (int p = 0; p < 4; ++p) {                 // B: 128x32 floats = 1024 b128
            const int row = lr + 32 * p;
            async_copy_b128(&Bs[buf][row][lc],
                            W + (size_t)(n0 + row) * K_DIM + kt + lc);
        }
    };

    // Prologue: stage 0, then wait for every wave's copies before first compute.
    issue_stage(0, 0);
    wait_async0();
    __syncthreads();

    int buf = 0;
    for (int kt = 0; kt < K_DIM; kt += BK) {
        // Overlap: start the DMA for the next tile into the other buffer.
        if (kt + BK < K_DIM) issue_stage(kt + BK, buf ^ 1);

        // ---- 8 WMMA k-steps of 4 on the current buffer ----
        #pragma unroll
        for (int ks = 0; ks < BK / 4; ++ks) {
            const int k0 = ks * 4 + 2 * lhalf;        // lane holds K = k0 + {0,1}
            v2f a0 = *(const v2f*)&As[buf][wm + lmod][k0];
            v2f a1 = *(const v2f*)&As[buf][wm + 16 + lmod][k0];
            // per-token fake quant (rintf == round-half-to-even, matches jnp.round)
            a0[0] = rintf(a0[0] * r0) * s0;  a0[1] = rintf(a0[1] * r0) * s0;
            a1[0] = rintf(a1[0] * r1) * s1;  a1[1] = rintf(a1[1] * r1) * s1;
            v2f b0 = *(const v2f*)&Bs[buf][wn + lmod][k0];      // lane: N=lmod, K=k0+{0,1}
            v2f b1 = *(const v2f*)&Bs[buf][wn + 16 + lmod][k0];
            acc[0][0] = __builtin_amdgcn_wmma_f32_16x16x4_f32(false, a0, false, b0,
                                                              (short)0, acc[0][0], false, false);
            acc[0][1] = __builtin_amdgcn_wmma_f32_16x16x4_f32(false, a0, false, b1,
                                                              (short)0, acc[0][1], false, false);
            acc[1][0] = __builtin_amdgcn_wmma_f32_16x16x4_f32(false, a1, false, b0,
                                                              (short)0, acc[1][0], false, false);
            acc[1][1] = __builtin_amdgcn_wmma_f32_16x16x4_f32(false, a1, false, b1,
                                                              (short)0, acc[1][1], false, false);
        }

        // Handoff: next tile's DMA complete + all waves done with this buffer.
        wait_async0();
        __syncthreads();
        buf ^= 1;
    }

    // ---- epilogue: C layout (16x16 f32): lane -> N=lmod, M=vgpr + 8*lhalf; fuse bias ----
    #pragma unroll
    for (int i = 0; i < 2; ++i) {
        #pragma unroll
        for (int j = 0; j < 2; ++j) {
            const int n = n0 + wn + 16 * j + lmod;
            const float bv = bias[n];
            #pragma unroll
            for (int v = 0; v < 8; ++v) {
                const int m = m0 + wm + 16 * i + 8 * lhalf + v;
                y[(size_t)m * N_DIM + n] = acc[i][j][v] + bv;
            }
        }
    }
}

// ---------------------------------------------------------------------------
// Kernel 3: per-token requant of y, in place
// ---------------------------------------------------------------------------
__global__ void requant_kernel(float* __restrict__ y) {
    const int row = blockIdx.x;
    float* yr = y + (size_t)row * N_DIM;
    __shared__ float red[256];
    float m = 0.0f;
    for (int i = threadIdx.x; i < N_DIM; i += 256) m = fmaxf(m, fabsf(yr[i]));
    red[threadIdx.x] = m;
    __syncthreads();
    #pragma unroll
    for (int s = 128; s > 0; s >>= 1) {
        if (threadIdx.x < s) red[threadIdx.x] = fmaxf(red[threadIdx.x], red[threadIdx.x + s]);
        __syncthreads();
    }
    const float sc  = fmaxf(red[0], 1e-5f) * (1.0f / 127.0f);
    const float inv = 1.0f / sc;
    for (int i = threadIdx.x; i < N_DIM; i += 256) yr[i] = rintf(yr[i] * inv) * sc;
}

// ---------------------------------------------------------------------------
extern "C" void kernel_launch(void* const* d_in, const int* in_sizes, int n_in,
                              void* d_out, int out_size, void* d_ws, size_t ws_size,
                              hipStream_t stream) {
    const float* x = (const float*)d_in[0];   // [4,2048,4096]
    const float* W = (const float*)d_in[1];   // [4096,4096]
    const float* b = (const float*)d_in[2];   // [4096]
    float* y       = (float*)d_out;           // [4,2048,4096]
    float* scales  = (float*)d_ws;            // 8192 floats (32 KB scratch)

    row_scale_kernel<<<M_TOK, 256, 0, stream>>>(x, scales);

    dim3 grid(N_DIM / BN, M_TOK / BM);
    gemm_wmma_f32<<<grid, 256, 0, stream>>>(x, scales, W, b, y);

    requant_kernel<<<M_TOK, 256, 0, stream>>>(y);
}